// SerialBasedFeatureFusion_18159121727685
// MI455X (gfx1250) — compile-verified
//
#include <hip/hip_runtime.h>
#include <cstdint>

#define N_ROWS 65536
#define N_FEAT 1024
#define NBINS  30
#define HPAD   31   // LDS stride per thread-histogram (odd -> spreads banks)

typedef __attribute__((ext_vector_type(4))) float v4f;  // native vector: OK for nontemporal builtin

// ---- order-preserving float <-> uint map (for u32 atomic min/max) ----------
__device__ __forceinline__ unsigned f2ord(float f) {
  unsigned u = __float_as_uint(f);
  return (u & 0x80000000u) ? ~u : (u | 0x80000000u);
}
__device__ __forceinline__ float ord2f(unsigned o) {
  unsigned u = (o & 0x80000000u) ? (o & 0x7FFFFFFFu) : ~o;
  return __uint_as_float(u);
}

// ---- 0: workspace init (must run every launch; harness doesn't re-poison) --
__global__ void kinit(unsigned* __restrict__ omin, unsigned* __restrict__ omax,
                      unsigned* __restrict__ hist) {
  const int i = blockIdx.x * 256 + threadIdx.x;
  if (i < 2 * N_FEAT) { omin[i] = 0xFFFFFFFFu; omax[i] = 0u; }
  if (i < 2 * N_FEAT * NBINS) hist[i] = 0u;
}

// ---- 1: per-column min/max (coalesced: thread == column) -------------------
__global__ __launch_bounds__(256) void kminmax(const float* __restrict__ a,
                                               const float* __restrict__ b,
                                               unsigned* __restrict__ omin,
                                               unsigned* __restrict__ omax,
                                               int rowsPer) {
  const int z = blockIdx.z;
  const float* __restrict__ x = z ? b : a;
  const int col = blockIdx.x * 256 + threadIdx.x;
  const size_t r0 = (size_t)blockIdx.y * rowsPer;
  const float* p = x + r0 * N_FEAT + col;
  float mn = p[0], mx = p[0];
  for (int r = 1; r < rowsPer; ++r) {
    float v = p[(size_t)r * N_FEAT];
    mn = fminf(mn, v);
    mx = fmaxf(mx, v);
  }
  const int gcol = z * N_FEAT + col;
  atomicMin(&omin[gcol], f2ord(mn));
  atomicMax(&omax[gcol], f2ord(mx));
}

// ---- 2: histogram via private LDS bins (ds_add_u32), merged with atomics ---
__global__ __launch_bounds__(256) void khist(const float* __restrict__ a,
                                             const float* __restrict__ b,
                                             const unsigned* __restrict__ omin,
                                             const unsigned* __restrict__ omax,
                                             unsigned* __restrict__ hist,
                                             int rowsPer) {
  __shared__ unsigned lh[256 * HPAD];   // ~31 KB of 320 KB LDS
  const int z = blockIdx.z;
  const float* __restrict__ x = z ? b : a;
  const int col = blockIdx.x * 256 + threadIdx.x;
  for (int i = threadIdx.x; i < 256 * HPAD; i += 256) lh[i] = 0u;
  __syncthreads();

  const int gcol = z * N_FEAT + col;
  const float mn   = ord2f(omin[gcol]);
  const float mx   = ord2f(omax[gcol]);
  const float rng  = mx - mn;
  const float safe = (rng > 0.0f) ? rng : 1.0f;

  const size_t r0 = (size_t)blockIdx.y * rowsPer;
  const float* p = x + r0 * N_FEAT + col;
  unsigned* mylh = lh + threadIdx.x * HPAD;

  for (int r = 0; r < rowsPer; ++r) {
    float v = p[(size_t)r * N_FEAT];
    // exact reference op order: ((v-mn)/safe)*30, truncate, clamp [0,29]
    float scaled = (v - mn) / safe;
    int bi = (int)(scaled * 30.0f);
    bi = bi < 0 ? 0 : (bi > 29 ? 29 : bi);
    atomicAdd(&mylh[bi], 1u);           // ds_add_u32, fire-and-forget
  }
  __syncthreads();

  unsigned* gh = hist + (size_t)gcol * NBINS;
#pragma unroll
  for (int bb = 0; bb < NBINS; ++bb) {
    unsigned c = mylh[bb];
    if (c) atomicAdd(&gh[bb], c);
  }
}

// ---- 3: entropy per column -------------------------------------------------
__global__ void kentropy(const unsigned* __restrict__ hist,
                         const unsigned* __restrict__ omin,
                         const unsigned* __restrict__ omax,
                         float* __restrict__ H) {
  const int i = blockIdx.x * 256 + threadIdx.x;
  if (i >= 2 * N_FEAT) return;
  const unsigned* h = hist + (size_t)i * NBINS;
  float tot = 0.0f;
#pragma unroll
  for (int bb = 0; bb < NBINS; ++bb) tot += (float)h[bb];
  float s = 0.0f;
#pragma unroll
  for (int bb = 0; bb < NBINS; ++bb) {
    float pr = (float)h[bb] / tot;
    s += pr * logf(pr + 1e-12f);
  }
  float rng = ord2f(omax[i]) - ord2f(omin[i]);
  H[i] = (rng > 0.0f) ? -s : 0.0f;
}

// ---- 4: stable descending argsort via rank-by-count; scatter perm ----------
__global__ __launch_bounds__(1024) void krank(const float* __restrict__ H,
                                              int* __restrict__ perm) {
  __shared__ float sh[N_FEAT];
  const int z = blockIdx.x;       // 0 = A, 1 = B
  const int f = threadIdx.x;
  sh[f] = H[z * N_FEAT + f];
  __syncthreads();
  const float hf = sh[f];
  int r = 0;
  for (int g = 0; g < N_FEAT; ++g) {
    float hg = sh[g];
    r += (hg > hf) || ((hg == hf) && (g < f));
  }
  perm[z * N_FEAT + r] = f;       // dest column r takes source column f
}

// ---- 5: column gather, rows staged in LDS via async-to-LDS DMA -------------
#define G_ROWS 32
__global__ __launch_bounds__(256) void kgather(const float* __restrict__ a,
                                               const float* __restrict__ b,
                                               const int* __restrict__ permG,
                                               float* __restrict__ out) {
  __shared__ float buf[2][2048];  // [a-row | b-row], double buffered (16 KB)
  __shared__ int   perm[2048];    // 8 KB
  const int t = threadIdx.x;
  for (int i = t; i < 2048; i += 256) perm[i] = permG[i];

  const size_t n0 = (size_t)blockIdx.x * G_ROWS;

  // prime buffer 0 with row n0 (each thread DMAs 16B of a-row + 16B of b-row)
  {
    const float* ga = a + n0 * N_FEAT + t * 4;
    const float* gb = b + n0 * N_FEAT + t * 4;
    unsigned la = (unsigned)(uintptr_t)&buf[0][t * 4];
    unsigned lb = (unsigned)(uintptr_t)&buf[0][1024 + t * 4];
    asm volatile("global_load_async_to_lds_b128 %0, %1, off" :: "v"(la), "v"(ga) : "memory");
    asm volatile("global_load_async_to_lds_b128 %0, %1, off" :: "v"(lb), "v"(gb) : "memory");
  }

  const int base = (t < 128) ? 0 : 1024;  // this thread's 8 outputs are all in one half
  for (int i = 0; i < G_ROWS; ++i) {
    const size_t n = n0 + i;
    if (i + 1 < G_ROWS) {
      // prefetch next row into the other buffer, then retire the older row's 2 ops
      const size_t n1 = n + 1;
      const float* ga = a + n1 * N_FEAT + t * 4;
      const float* gb = b + n1 * N_FEAT + t * 4;
      unsigned la = (unsigned)(uintptr_t)&buf[(i + 1) & 1][t * 4];
      unsigned lb = (unsigned)(uintptr_t)&buf[(i + 1) & 1][1024 + t * 4];
      asm volatile("global_load_async_to_lds_b128 %0, %1, off" :: "v"(la), "v"(ga) : "memory");
      asm volatile("global_load_async_to_lds_b128 %0, %1, off" :: "v"(lb), "v"(gb) : "memory");
      asm volatile("s_wait_asynccnt 2" ::: "memory");
    } else {
      asm volatile("s_wait_asynccnt 0" ::: "memory");
    }
    __syncthreads();  // all waves' DMA for row i has landed in LDS

    const float* rb = buf[i & 1];
    const int j0 = t * 8;
    float v[8];
#pragma unroll
    for (int k = 0; k < 8; ++k) v[k] = rb[base + perm[j0 + k]];
    v4f o0 = {v[0], v[1], v[2], v[3]};
    v4f o1 = {v[4], v[5], v[6], v[7]};
    v4f* op = (v4f*)(out + n * 2048 + j0);
    __builtin_nontemporal_store(o0, op);
    __builtin_nontemporal_store(o1, op + 1);

    __syncthreads();  // buffer (i&1) free for the DMA issued next iteration
  }
}

extern "C" void kernel_launch(void* const* d_in, const int* in_sizes, int n_in,
                              void* d_out, int out_size, void* d_ws, size_t ws_size,
                              hipStream_t stream) {
  const float* a = (const float*)d_in[0];
  const float* b = (const float*)d_in[1];
  float* out = (float*)d_out;

  // workspace layout (all u32-sized): 2048 + 2048 + 61440 + 2048 + 2048 = 272 KB
  unsigned* omin = (unsigned*)d_ws;
  unsigned* omax = omin + 2 * N_FEAT;
  unsigned* hist = omax + 2 * N_FEAT;
  float*    H    = (float*)(hist + 2 * N_FEAT * NBINS);
  int*      perm = (int*)(H + 2 * N_FEAT);

  kinit<<<240, 256, 0, stream>>>(omin, omax, hist);
  {
    dim3 g(N_FEAT / 256, N_ROWS / 256, 2);
    kminmax<<<g, 256, 0, stream>>>(a, b, omin, omax, 256);
  }
  {
    dim3 g(N_FEAT / 256, N_ROWS / 512, 2);
    khist<<<g, 256, 0, stream>>>(a, b, omin, omax, hist, 512);
  }
  kentropy<<<8, 256, 0, stream>>>(hist, omin, omax, H);
  krank<<<2, 1024, 0, stream>>>(H, perm);
  kgather<<<N_ROWS / G_ROWS, 256, 0, stream>>>(a, b, perm, out);
}